// GNNBaseline_46729244181042
// MI455X (gfx1250) — compile-verified
//
#include <hip/hip_runtime.h>
#include <hip/hip_bf16.h>
#include <stdint.h>

typedef __attribute__((ext_vector_type(2))) float v2f;
typedef __attribute__((ext_vector_type(8))) float v8f;

#define HDIM 64

// ---------------- utility kernels ----------------

__global__ void fill_kernel(float* __restrict__ p, float v, long n) {
  long i = (long)blockIdx.x * blockDim.x + threadIdx.x;
  if (i < n) p[i] = v;
}

__global__ void degree_kernel(const int* __restrict__ dst, float* __restrict__ deg, int E) {
  int e = blockIdx.x * blockDim.x + threadIdx.x;
  if (e < E) atomicAdd(&deg[dst[e]], 1.0f);
}

__global__ void rsqrt_kernel(float* __restrict__ d, int N) {
  int i = blockIdx.x * blockDim.x + threadIdx.x;
  if (i < N) d[i] = rsqrtf(d[i]);   // deg >= 1 always (self loop)
}

// ---------------- WMMA GEMM: T[N x 64] = A[N x K] @ W[K x 64] ----------------
// f32 WMMA 16x16x4. Layouts per CDNA5 ISA:
//   A 16x4 : lane r=lane&15 -> M=r ; kh=lane>>4 ; VGPR0=K(2*kh), VGPR1=K(2*kh+1)
//   B 4x16 : lane r -> N=r ; VGPR0=K(2*kh), VGPR1=K(2*kh+1)
//   C 16x16: VGPR j -> M = j + 8*kh, N = r
// W is staged into LDS pair-interleaved: sW2[kp*128 + n*2 + (k&1)] = W[k*64+n]
// so a B fragment is ONE aligned ds_load_b64 into an even VGPR pair (no repacking).
template <int K>
__global__ __launch_bounds__(256) void gemm_wmma_kernel(
    const float* __restrict__ A, const float* __restrict__ W,
    float* __restrict__ T, int nrows) {
  __shared__ float sW2[K * HDIM];
  const int tid = threadIdx.x;
  // stage: thread handles one (k-pair, n): read W[2kp][n], W[2kp+1][n], store contiguous
  for (int i = tid; i < (K / 2) * HDIM; i += 256) {
    int kp = i >> 6;          // k-pair index
    int n  = i & 63;
    float2 w2;
    w2.x = W[(2 * kp)     * HDIM + n];
    w2.y = W[(2 * kp + 1) * HDIM + n];
    *(float2*)&sW2[kp * 2 * HDIM + 2 * n] = w2;
  }
  __syncthreads();

  const int wave = tid >> 5;
  const int lane = tid & 31;
  const int tile = blockIdx.x * 8 + wave;
  const int row0 = tile * 16;
  if (row0 >= nrows) return;          // wave-uniform: EXEC stays all-ones for WMMA

  const int r  = lane & 15;
  const int kh = lane >> 4;           // 0: K pair {0,1} / M 0..7 ; 1: K pair {2,3} / M 8..15

  int rowa = row0 + r; if (rowa >= nrows) rowa = nrows - 1;
  const float* __restrict__ arow = A + (size_t)rowa * K + 2 * kh;

  v8f acc0 = {}, acc1 = {}, acc2 = {}, acc3 = {};

#pragma unroll
  for (int kk = 0; kk < K; kk += 4) {
    v2f a;
    a.x = arow[kk];
    a.y = arow[kk + 1];
    // B pair row for this half-wave: kp = kk/2 + kh ; column n = nt*16 + r
    const float* bp = &sW2[(kk / 2 + kh) * 2 * HDIM + 2 * r];
    v2f b;
    b = *(const v2f*)(bp + 0);
    acc0 = __builtin_amdgcn_wmma_f32_16x16x4_f32(false, a, false, b, (short)0, acc0, false, false);
    b = *(const v2f*)(bp + 32);
    acc1 = __builtin_amdgcn_wmma_f32_16x16x4_f32(false, a, false, b, (short)0, acc1, false, false);
    b = *(const v2f*)(bp + 64);
    acc2 = __builtin_amdgcn_wmma_f32_16x16x4_f32(false, a, false, b, (short)0, acc2, false, false);
    b = *(const v2f*)(bp + 96);
    acc3 = __builtin_amdgcn_wmma_f32_16x16x4_f32(false, a, false, b, (short)0, acc3, false, false);
  }

  float* __restrict__ out = T + (size_t)row0 * HDIM + r;  // column r base
#pragma unroll
  for (int j = 0; j < 8; ++j) {
    const int m = j + 8 * kh;
    if (row0 + m < nrows) {
      const size_t ro = (size_t)m * HDIM;
      out[ro +  0] = acc0[j];
      out[ro + 16] = acc1[j];
      out[ro + 32] = acc2[j];
      out[ro + 48] = acc3[j];
    }
  }
}

// ---------------- edge scatter: agg[dst] += T[src] * (dinv[src]*dinv[dst]) ----------------
// one wave per edge; each lane moves a float2 (64 features / 32 lanes).
// edge/index loads forced scalar (wave-uniform) so vector-memory slots go to the atomics.
__global__ void scatter_kernel(const float* __restrict__ T, const int* __restrict__ src,
                               const int* __restrict__ dst, const float* __restrict__ dinv,
                               float* __restrict__ agg, int E) {
  long t = (long)blockIdx.x * blockDim.x + threadIdx.x;
  long e = t >> 5;
  if (e >= E) return;
  int eu = __builtin_amdgcn_readfirstlane((int)e);   // wave-uniform edge id -> SGPR
  int p = (int)(t & 31);
  int s = src[eu], d = dst[eu];                      // s_load
  float nm = dinv[s] * dinv[d];
  const float2* tv = (const float2*)(T + (size_t)s * HDIM);
  float2 v = tv[p];
  float* ap = agg + (size_t)d * HDIM + 2 * p;
  atomicAdd(ap,     v.x * nm);   // resolves in 192MB L2 (agg fits)
  atomicAdd(ap + 1, v.y * nm);
}

// ---------------- epilogue: H = (agg + T*dinv^2 + b) [relu] ----------------
__global__ void epilogue_kernel(const float* __restrict__ agg, const float* __restrict__ T,
                                const float* __restrict__ dinv, const float* __restrict__ bias,
                                float* __restrict__ H, long N, int do_relu) {
  long t = (long)blockIdx.x * blockDim.x + threadIdx.x;
  if (t >= N * HDIM) return;
  long node = t >> 6;
  int  f    = (int)(t & 63);
  float sn = dinv[node]; sn *= sn;                 // self-loop norm = 1/deg
  float v = agg[t] + T[t] * sn + bias[f];
  H[t] = do_relu ? fmaxf(v, 0.0f) : v;
}

// ---------------- pool: per-node dot(H[node], Wl) scattered into graph sums ----------------
__global__ void pool_kernel(const float* __restrict__ H, const float* __restrict__ Wl,
                            const int* __restrict__ batch, float* __restrict__ gsum,
                            float* __restrict__ gcnt, int N) {
  long t = (long)blockIdx.x * blockDim.x + threadIdx.x;
  int node = (int)(t >> 5);
  int lane = (int)(t & 31);
  if (node >= N) return;
  const float2* hv = (const float2*)(H + (size_t)node * HDIM);
  const float2* wv = (const float2*)Wl;
  float2 h = hv[lane], w = wv[lane];
  float part = h.x * w.x + h.y * w.y;
#pragma unroll
  for (int off = 16; off > 0; off >>= 1)
    part += __shfl_xor(part, off, 32);             // wave32 reduction
  if (lane == 0) {
    int g = batch[node];
    atomicAdd(&gsum[g], part);
    atomicAdd(&gcnt[g], 1.0f);
  }
}

__global__ void final_kernel(const float* __restrict__ gsum, const float* __restrict__ gcnt,
                             const float* __restrict__ bl, float* __restrict__ out, int G) {
  int g = blockIdx.x * blockDim.x + threadIdx.x;
  if (g < G) out[g] = gsum[g] / fmaxf(gcnt[g], 1.0f) + bl[0];
}

// ---------------- launch ----------------

static inline int cdiv_l(long a, long b) { return (int)((a + b - 1) / b); }

extern "C" void kernel_launch(void* const* d_in, const int* in_sizes, int n_in,
                              void* d_out, int out_size, void* d_ws, size_t ws_size,
                              hipStream_t stream) {
  const float* x     = (const float*)d_in[0];
  const int*   ei    = (const int*)  d_in[1];
  const int*   batch = (const int*)  d_in[2];
  const float* W1    = (const float*)d_in[3];
  const float* b1    = (const float*)d_in[4];
  const float* W2    = (const float*)d_in[5];
  const float* b2    = (const float*)d_in[6];
  const float* W3    = (const float*)d_in[7];
  const float* b3    = (const float*)d_in[8];
  const float* Wl    = (const float*)d_in[9];
  const float* bl    = (const float*)d_in[10];
  float* out = (float*)d_out;

  const int  Hd  = in_sizes[4];                 // 64
  const int  Fin = in_sizes[3] / Hd;            // 128
  const long N   = (long)in_sizes[0] / Fin;     // 100000
  const long E   = (long)in_sizes[1] / 2;       // 1600000
  const int  G   = out_size;                    // 1000

  const int* srcp = ei;
  const int* dstp = ei + E;

  // workspace carve-out (~77 MB)
  float* dinv = (float*)d_ws;
  float* T    = dinv + N;
  float* agg  = T    + N * HDIM;
  float* Hbuf = agg  + N * HDIM;
  float* gsum = Hbuf + N * HDIM;
  float* gcnt = gsum + G;

  const int BT = 256;
  const long NH = N * HDIM;
  const int tiles   = cdiv_l(N, 16);
  const int gblocks = cdiv_l(tiles, 8);

  // normalization coefficients
  fill_kernel   <<<cdiv_l(N, BT), BT, 0, stream>>>(dinv, 1.0f, N);   // self-loop degree
  degree_kernel <<<cdiv_l(E, BT), BT, 0, stream>>>(dstp, dinv, (int)E);
  rsqrt_kernel  <<<cdiv_l(N, BT), BT, 0, stream>>>(dinv, (int)N);

  // layer 1: 128 -> 64, relu
  gemm_wmma_kernel<128><<<gblocks, BT, 0, stream>>>(x, W1, T, (int)N);
  fill_kernel    <<<cdiv_l(NH, BT), BT, 0, stream>>>(agg, 0.0f, NH);
  scatter_kernel <<<cdiv_l(E * 32, BT), BT, 0, stream>>>(T, srcp, dstp, dinv, agg, (int)E);
  epilogue_kernel<<<cdiv_l(NH, BT), BT, 0, stream>>>(agg, T, dinv, b1, Hbuf, N, 1);

  // layer 2: 64 -> 64, relu
  gemm_wmma_kernel<64><<<gblocks, BT, 0, stream>>>(Hbuf, W2, T, (int)N);
  fill_kernel    <<<cdiv_l(NH, BT), BT, 0, stream>>>(agg, 0.0f, NH);
  scatter_kernel <<<cdiv_l(E * 32, BT), BT, 0, stream>>>(T, srcp, dstp, dinv, agg, (int)E);
  epilogue_kernel<<<cdiv_l(NH, BT), BT, 0, stream>>>(agg, T, dinv, b2, Hbuf, N, 1);

  // layer 3: 64 -> 64, no relu
  gemm_wmma_kernel<64><<<gblocks, BT, 0, stream>>>(Hbuf, W3, T, (int)N);
  fill_kernel    <<<cdiv_l(NH, BT), BT, 0, stream>>>(agg, 0.0f, NH);
  scatter_kernel <<<cdiv_l(E * 32, BT), BT, 0, stream>>>(T, srcp, dstp, dinv, agg, (int)E);
  epilogue_kernel<<<cdiv_l(NH, BT), BT, 0, stream>>>(agg, T, dinv, b3, Hbuf, N, 0);

  // mean pool + linear head (commuted: dot first, then scatter-mean)
  fill_kernel <<<cdiv_l(G, BT), BT, 0, stream>>>(gsum, 0.0f, G);
  fill_kernel <<<cdiv_l(G, BT), BT, 0, stream>>>(gcnt, 0.0f, G);
  pool_kernel <<<cdiv_l(N * 32, BT), BT, 0, stream>>>(Hbuf, Wl, batch, gsum, gcnt, (int)N);
  final_kernel<<<cdiv_l(G, BT), BT, 0, stream>>>(gsum, gcnt, bl, out, G);
}